// Layer_1331439861945
// MI455X (gfx1250) — compile-verified
//
#include <hip/hip_runtime.h>

typedef __attribute__((ext_vector_type(2))) float v2f;
typedef __attribute__((ext_vector_type(8))) float v8f;

#define MT 65536   // M = B*T
#define KC 256     // K = C
#define NH 256     // N = H

// ---------------------------------------------------------------------------
// prep_weights: Wf = (Wi + Wh), stored in WMMA B-fragment order.
// float2 index idx = ((s*16 + t)*32 + lane):
//   n = t*16 + (lane&15); k = s*4 + ((lane>>4)<<1); .x -> W[n][k], .y -> W[n][k+1]
// Matches ISA 16x16x4 B layout (VGPR0: K=0 / K=2, VGPR1: K=1 / K=3); the LDS
// read becomes a bank-conflict-free ds_load_b64.
// ---------------------------------------------------------------------------
__global__ __launch_bounds__(256) void prep_weights(const float* __restrict__ Wi,
                                                    const float* __restrict__ Wh,
                                                    float* __restrict__ Wf) {
    int idx  = blockIdx.x * 256 + threadIdx.x;   // 0 .. 32767 (float2 index)
    int lane = idx & 31;
    int st   = idx >> 5;
    int s    = st >> 4;
    int t    = st & 15;
    int k    = s * 4 + ((lane >> 4) << 1);
    int n    = t * 16 + (lane & 15);
    v2f w;
    w.x = Wi[n * KC + k]     + Wh[n * KC + k];
    w.y = Wi[n * KC + k + 1] + Wh[n * KC + k + 1];
    ((v2f*)Wf)[idx] = w;
}

// Branchless fp32 tanh using hardware transcendentals:
//   tanh(x) = 1 - 2/(exp(2x)+1) ; exp(2x) = exp2(x * 2*log2(e))
// v_exp_f32 + v_rcp_f32, no EXEC divergence; saturates correctly at +-inf.
__device__ __forceinline__ float fast_tanh(float v) {
    const float TWO_LOG2E = 2.885390081777926815f;   // 2*log2(e)
    float e = __builtin_amdgcn_exp2f(v * TWO_LOG2E); // v_exp_f32 (2^x)
    float r = __builtin_amdgcn_rcpf(e + 1.0f);       // v_rcp_f32
    return 1.0f - 2.0f * r;                          // fma
}

// ---------------------------------------------------------------------------
// gemm_tanh: hidden = tanh(X (M x K) * W^T + b), fp32 WMMA.
// 256 threads = 8 waves. Wave w computes M-tile (blockIdx.x*8 + w), all of N.
// Full weight matrix (256 KB) staged once into LDS (320 KB/WGP on CDNA5).
// ---------------------------------------------------------------------------
__global__ __launch_bounds__(256, 1) void gemm_tanh(const float* __restrict__ x,
                                                    const float* __restrict__ Wf,
                                                    const float* __restrict__ bi,
                                                    const float* __restrict__ bh,
                                                    float* __restrict__ out,
                                                    float* __restrict__ hlast) {
    __shared__ float ldsW[KC * NH];              // 256 KB, fragment-ordered

    const int tid = threadIdx.x;

    // Cooperative stage of the full weight matrix: 16384 x b128, coalesced.
    for (int i = tid; i < (KC * NH) / 4; i += 256) {
        ((float4*)ldsW)[i] = ((const float4*)Wf)[i];
    }
    __syncthreads();

    const int lane  = tid & 31;
    const int wave  = tid >> 5;
    const int mtile = blockIdx.x * 8 + wave;

    // A-fragment addressing (ISA 7.12.2, 32-bit A 16x4):
    //   lane<16 : M=lane,    K = s*4 + {0,1}
    //   lane>=16: M=lane-16, K = s*4 + {2,3}
    const int arow  = mtile * 16 + (lane & 15);
    const float* xa = x + (size_t)arow * KC + ((lane >> 4) << 1);

    v8f acc[16];
    const v8f zero = {0.f, 0.f, 0.f, 0.f, 0.f, 0.f, 0.f, 0.f};
#pragma unroll
    for (int t = 0; t < 16; ++t) acc[t] = zero;

    const v2f* bfr = ((const v2f*)ldsW) + lane;

    for (int s = 0; s < 64; ++s) {               // K in steps of 4
        v2f a = *(const v2f*)(xa + s * 4);
#pragma unroll
        for (int t = 0; t < 16; ++t) {           // 16 N-tiles of 16
            v2f b = bfr[(s * 16 + t) * 32];      // conflict-free ds_load_b64
            acc[t] = __builtin_amdgcn_wmma_f32_16x16x4_f32(
                /*neg_a=*/false, a, /*neg_b=*/false, b,
                /*c_mod=*/(short)0, acc[t], /*reuse_a=*/false, /*reuse_b=*/false);
        }
    }

    // Epilogue. C/D layout: VGPR v, lane l -> M = v + 8*(l>=16), N = l%16.
    const int nb = lane & 15;
    const int m0 = mtile * 16 + (lane >> 4) * 8;
#pragma unroll
    for (int t = 0; t < 16; ++t) {
        const int n    = t * 16 + nb;
        const float bb = bi[n] + bh[n];
#pragma unroll
        for (int v = 0; v < 8; ++v) {
            const int m   = m0 + v;
            const float h = fast_tanh(acc[t][v] + bb);
            out[(size_t)m * NH + n] = h;
            if ((m & 4095) == 4095) {            // t == T-1 row -> h_last
                hlast[(size_t)(m >> 12) * NH + n] = h;
            }
        }
    }
}

extern "C" void kernel_launch(void* const* d_in, const int* in_sizes, int n_in,
                              void* d_out, int out_size, void* d_ws, size_t ws_size,
                              hipStream_t stream) {
    const float* x  = (const float*)d_in[0];   // (16, 4096, 256)
    const float* Wi = (const float*)d_in[1];   // (256, 256)
    const float* bi = (const float*)d_in[2];   // (256,)
    const float* Wh = (const float*)d_in[3];   // (256, 256)
    const float* bh = (const float*)d_in[4];   // (256,)

    float* out   = (float*)d_out;              // hidden_states, 16*4096*256
    float* hlast = out + (size_t)MT * NH;      // h_last, 16*256

    float* Wf = (float*)d_ws;                  // 256 KB swizzled weights

    prep_weights<<<128, 256, 0, stream>>>(Wi, Wh, Wf);
    gemm_tanh<<<MT / (8 * 16), 256, 0, stream>>>(x, Wf, bi, bh, out, hlast);
}